// DisparityFusion_58660663328884
// MI455X (gfx1250) — compile-verified
//
#include <hip/hip_runtime.h>
#include <hip/hip_bf16.h>

typedef __attribute__((ext_vector_type(2))) float v2f;
typedef __attribute__((ext_vector_type(8))) float v8f;

#define DMAX 192
#define BATCH 2
#define HH 256
#define WW 512
#define HW (HH * WW)

// -------------------------------------------------------------------------
// Kernel 1: per-pixel online softmax over the disparity axis + expectation.
// One thread per (b, branch, h, w). Lanes cover consecutive w -> every
// global_load_b32 is a fully coalesced 128B wave transaction. Single pass
// (online max/rescale) so each of the 604MB is read exactly once: HBM-bound.
// -------------------------------------------------------------------------
__global__ void df_softmax_regress(const float* __restrict__ c0,
                                   const float* __restrict__ c1,
                                   const float* __restrict__ c2,
                                   float* __restrict__ disp) {
    int pix = blockIdx.x * blockDim.x + threadIdx.x;   // 0 .. HW-1
    int bbr = blockIdx.y;                              // 0 .. 5  (b*3 + br)
    int br  = bbr % 3;
    const float* base = (br == 0) ? c0 : ((br == 1) ? c1 : c2);
    const float* src  = base + (size_t)(bbr / 3) * DMAX * HW + pix;

    float m = -1e30f, s = 0.f, t = 0.f;
    for (int d0 = 0; d0 < DMAX; d0 += 8) {
        float x[8];
#pragma unroll
        for (int j = 0; j < 8; ++j)            // 8 loads in flight per thread
            x[j] = src[(size_t)(d0 + j) * HW];
#pragma unroll
        for (int j = 0; j < 8; ++j) {
            float nm = fmaxf(m, x[j]);
            float r  = __expf(m - nm);         // rescale old partial sums
            float e  = __expf(x[j] - nm);
            s = s * r + e;
            t = t * r + e * (float)(d0 + j);
            m = nm;
        }
    }
    disp[(size_t)bbr * HW + pix] = t / s;      // E[d] under softmax
}

// -------------------------------------------------------------------------
// Kernel 2: fused conv3x3(1->9)+BN+ReLU (x3 branches), CSPN gate
// normalization, and affinity-weighted 3x3 propagation.
// The conv is an im2col GEMM executed with V_WMMA_F32_16X16X4_F32:
//   M = 16 pixels, N = 16 (channels 0..8 valid), K = 9 padded to 12.
// Block = 256 threads (8 waves); tile = 8 rows x 32 cols; each wave owns one
// row and runs two 16-pixel WMMA groups.
// -------------------------------------------------------------------------
__global__ void df_fusion_wmma(const float* __restrict__ disp,
                               const float* __restrict__ Wc1, const float* __restrict__ Gc1, const float* __restrict__ Bc1,
                               const float* __restrict__ Wc2, const float* __restrict__ Gc2, const float* __restrict__ Bc2,
                               const float* __restrict__ Wc3, const float* __restrict__ Gc3, const float* __restrict__ Bc3,
                               float* __restrict__ out) {
    __shared__ float sdisp[3][10][34];     // disp tile + 1-halo, zero padded
    __shared__ float saff[8][27][33];      // per-wave conv results (stride 33: no bank conflicts)

    const int b      = blockIdx.z;
    const int tileH0 = blockIdx.y * 8;
    const int tileW0 = blockIdx.x * 32;
    const int wave   = threadIdx.x >> 5;
    const int lane   = threadIdx.x & 31;
    const int half   = lane >> 4;          // K-half of the wave (ISA f32 A/B layout)
    const int mloc   = lane & 15;          // pixel row (A) / channel col (B)

    // ---- stage disp tile (3 branches, +halo) into LDS, zero outside image
    for (int i = threadIdx.x; i < 3 * 10 * 34; i += 256) {
        int br  = i / 340;
        int rem = i % 340;
        int rr  = rem / 34, cc = rem % 34;
        int h = tileH0 + rr - 1;
        int w = tileW0 + cc - 1;
        float v = 0.f;
        if (h >= 0 && h < HH && w >= 0 && w < WW)
            v = disp[((size_t)(b * 3 + br) * HH + h) * WW + w];
        sdisp[br][rr][cc] = v;
    }
    __syncthreads();

    // ---- per-lane B-matrix fragments: B[k][c] = W[c*9+k], zero-padded
    const float* Wb[3] = {Wc1, Wc2, Wc3};
    const float* Gb[3] = {Gc1, Gc2, Gc3};
    const float* Bb[3] = {Bc1, Bc2, Bc3};
    float bf[3][3][2];
#pragma unroll
    for (int br = 0; br < 3; ++br)
#pragma unroll
        for (int kc = 0; kc < 3; ++kc)
#pragma unroll
            for (int j = 0; j < 2; ++j) {
                int k = kc * 4 + half * 2 + j;
                bf[br][kc][j] = (mloc < 9 && k < 9) ? Wb[br][mloc * 9 + k] : 0.f;
            }

    // ---- conv phase: two 16-pixel groups per wave, 3 branches each
    for (int g = 0; g < 2; ++g) {
        const int g16 = g * 16;
#pragma unroll
        for (int br = 0; br < 3; ++br) {
            v8f acc = {0.f, 0.f, 0.f, 0.f, 0.f, 0.f, 0.f, 0.f};
#pragma unroll
            for (int kc = 0; kc < 3; ++kc) {
                // A fragment: patch values, lane-half selects K pair (ISA 16x4 f32 layout)
                int k0 = kc * 4 + half * 2;
                float a0 = 0.f, a1 = 0.f;
                if (k0 <= 8)
                    a0 = sdisp[br][wave + k0 / 3][g16 + mloc + (k0 % 3)];
                if (k0 + 1 <= 8)
                    a1 = sdisp[br][wave + (k0 + 1) / 3][g16 + mloc + ((k0 + 1) % 3)];
                v2f A = {a0, a1};
                v2f Bv = {bf[br][kc][0], bf[br][kc][1]};
                acc = __builtin_amdgcn_wmma_f32_16x16x4_f32(
                    /*neg_a=*/false, A, /*neg_b=*/false, Bv,
                    /*c_mod=*/(short)0, acc, /*reuse_a=*/false, /*reuse_b=*/false);
            }
            // D layout: lane = channel (N), VGPR r = pixel M = r + half*8.
            // Apply BN (gamma/beta) + ReLU and park in per-wave LDS.
            if (mloc < 9) {
                float gch = Gb[br][mloc];
                float bch = Bb[br][mloc];
#pragma unroll
                for (int r = 0; r < 8; ++r) {
                    float y = fmaxf(acc[r] * gch + bch, 0.f);
                    saff[wave][br * 9 + mloc][g16 + half * 8 + r] = y;
                }
            }
        }
    }
    // saff is private to this wave; LDS ops within a wave are in-order.

    // ---- gate normalization + affinity-weighted 3x3x3 propagation,
    //      one pixel per lane (row = wave, col = lane)
    float affv[27];
    float ssum = 0.f;
#pragma unroll
    for (int i = 0; i < 27; ++i) {
        float a = saff[wave][i][lane];
        affv[i] = a;
        ssum += fabsf(a);
    }
    float inv = 1.f / (ssum + 1e-8f);

    float o = 0.f;
#pragma unroll
    for (int br = 0; br < 3; ++br)
#pragma unroll
        for (int k = 0; k < 9; ++k)
            o += affv[br * 9 + k] * sdisp[br][wave + k / 3][lane + (k % 3)];

    int h = tileH0 + wave;
    int w = tileW0 + lane;
    out[((size_t)b * HH + h) * WW + w] = o * inv;
}

extern "C" void kernel_launch(void* const* d_in, const int* in_sizes, int n_in,
                              void* d_out, int out_size, void* d_ws, size_t ws_size,
                              hipStream_t stream) {
    const float* out1 = (const float*)d_in[0];
    const float* out2 = (const float*)d_in[1];
    const float* out3 = (const float*)d_in[2];
    const float* W1 = (const float*)d_in[3];
    const float* g1 = (const float*)d_in[4];
    const float* b1 = (const float*)d_in[5];
    const float* W2 = (const float*)d_in[6];
    const float* g2 = (const float*)d_in[7];
    const float* b2 = (const float*)d_in[8];
    const float* W3 = (const float*)d_in[9];
    const float* g3 = (const float*)d_in[10];
    const float* b3 = (const float*)d_in[11];

    float* disp = (float*)d_ws;   // [B=2][branch=3][H][W] = 3 MB scratch

    dim3 blk1(256);
    dim3 grd1(HW / 256, BATCH * 3);
    df_softmax_regress<<<grd1, blk1, 0, stream>>>(out1, out2, out3, disp);

    dim3 blk2(256);
    dim3 grd2(WW / 32, HH / 8, BATCH);
    df_fusion_wmma<<<grd2, blk2, 0, stream>>>(disp,
                                              W1, g1, b1,
                                              W2, g2, b2,
                                              W3, g3, b3,
                                              (float*)d_out);
}